// ACSF_71116068487423
// MI455X (gfx1250) — compile-verified
//
#include <hip/hip_runtime.h>
#include <hip/hip_bf16.h>

// CDNA5 / gfx1250 ACSF kernel.
// G4 block computed as per-atom f32 matmul D = A^T(18xT) * B(Tx8) using
// V_WMMA_F32_16X16X4_F32 (full-precision WMMA), one wave32 per atom.

typedef float v2f __attribute__((ext_vector_type(2)));
typedef float v8f __attribute__((ext_vector_type(8)));

#define NDESC 156
#define RCUT 5.0f
#define INV_RC2 0.04f            // 1/25
#define PI_OVER_RC 0.6283185307179586f

__device__ __forceinline__ float fcut(float r) {
    return (r < RCUT) ? 0.5f * (__cosf(PI_OVER_RC * r) + 1.0f) : 0.0f;
}

// ---------------------------------------------------------------- utility
__global__ void zero_kernel(float* __restrict__ out, int n) {
    int i = blockIdx.x * blockDim.x + threadIdx.x;
    if (i < n) out[i] = 0.0f;
}

__global__ void norm_kernel(float* __restrict__ out,
                            const float* __restrict__ mean,
                            const float* __restrict__ stdv, int n) {
    int i = blockIdx.x * blockDim.x + threadIdx.x;
    if (i >= n) return;
    int d = i % NDESC;
    out[i] = (out[i] - mean[d]) / stdv[d];
}

// ---------------------------------------------------------------- G2 (radial)
// One thread per edge; atomic f32 adds into columns [0,12) of the center atom.
__global__ void g2_kernel(const float* __restrict__ pos,
                          const float* __restrict__ cell,
                          const float* __restrict__ eshift,
                          const int* __restrict__ z,
                          const int* __restrict__ batch,
                          const int* __restrict__ eidx,
                          int E, float* __restrict__ out) {
    int e = blockIdx.x * blockDim.x + threadIdx.x;
    if (e >= E) return;
    const float eta2[6] = {0.01f, 0.05f, 1.1f, 1.9f, 2.0f, 9.0f};

    int i2 = eidx[e];       // edge_index[0] (segment key in reference)
    int j2 = eidx[E + e];   // edge_index[1]
    int cb = batch[i2] * 9;
    float s0 = eshift[e * 3 + 0], s1 = eshift[e * 3 + 1], s2 = eshift[e * 3 + 2];
    float sx = s0 * cell[cb + 0] + s1 * cell[cb + 3] + s2 * cell[cb + 6];
    float sy = s0 * cell[cb + 1] + s1 * cell[cb + 4] + s2 * cell[cb + 7];
    float sz = s0 * cell[cb + 2] + s1 * cell[cb + 5] + s2 * cell[cb + 8];
    float vx = pos[j2 * 3 + 0] - pos[i2 * 3 + 0] + sx;
    float vy = pos[j2 * 3 + 1] - pos[i2 * 3 + 1] + sy;
    float vz = pos[j2 * 3 + 2] - pos[i2 * 3 + 2] + sz;
    float r2 = vx * vx + vy * vy + vz * vz;
    float r  = sqrtf(r2);
    float fc = fcut(r);

    int zj = z[j2];
    int spb = (zj == 1) ? 0 : ((zj == 8) ? 6 : -1);
    if (spb < 0) return;
    float* orow = out + (size_t)i2 * NDESC + spb;
#pragma unroll
    for (int k = 0; k < 6; ++k)
        atomicAdd(&orow[k], __expf(-eta2[k] * r2 * INV_RC2) * fc);
}

// ---------------------------------------------------------------- G4 (angular)
// One wave32 per atom. Per chunk of 32 triplets we stage, in LDS:
//   rad[t][e]  = exp(-eta4[e]*(rij^2+rik^2+rjk^2)/rc^2)          (6 floats)
//   bang[t][q] = fc3 * c * (1 +/- cos)^zeta, zeta=1,2,4,8, c=1,1/2,1/8,1/64              (8 floats)
//   cls[t]     = species class of (zj,zk): 0=(1,1) 1=(8,8) 2=mixed
// Then D[s*6+e][q] = sum_t A[row][t]*B[t][q] via v_wmma_f32_16x16x4_f32:
//   wmma0 rows 0..5 -> s=0, rows 6..11 -> s=1 ; wmma1 rows 0..5 -> s=2.
__global__ __launch_bounds__(32) void g4_kernel(
    const float* __restrict__ pos, const float* __restrict__ cell,
    const float* __restrict__ est, const int* __restrict__ z,
    const int* __restrict__ batch,
    const int* __restrict__ idx_i, const int* __restrict__ idx_j,
    const int* __restrict__ idx_k, int n_tri, float* __restrict__ out) {

    const int a    = blockIdx.x;
    const int lane = threadIdx.x;
    const int hf   = lane >> 4;   // 0: K=0,1 ; 1: K=2,3
    const int lm   = lane & 15;   // matrix row (A) / column (B,D)

    __shared__ float s_rad[32][6];
    __shared__ float s_bang[32][8];
    __shared__ int   s_cls[32];

    const float eta4[6] = {0.01f, 0.1f, 0.5f, 1.1f, 1.5f, 2.5f};

    // wave-uniform binary search: [tstart, tend) = triplets with idx_i == a
    int lo = 0, hi = n_tri;
    while (lo < hi) { int m = (lo + hi) >> 1; if (idx_i[m] <  a) lo = m + 1; else hi = m; }
    const int tstart = lo;
    hi = n_tri;
    while (lo < hi) { int m = (lo + hi) >> 1; if (idx_i[m] <= a) lo = m + 1; else hi = m; }
    const int tend = lo;

    v8f c0 = {}; // species blocks 0,1
    v8f c1 = {}; // species block 2 (mixed)

    const float xi = pos[a * 3 + 0];
    const float yi = pos[a * 3 + 1];
    const float zi = pos[a * 3 + 2];
    const int   cb = batch[a] * 9;

    const int nt  = tend - tstart;
    const int nch = (nt + 31) >> 5;

    // precomputed per-lane fragment row selectors (uniform across chunks)
    const int  srow0  = (lm < 12) ? (lm / 6) : 0;     // species of wmma0 row
    const int  erow0  = (lm < 12) ? (lm - srow0 * 6) : 0;
    const bool rowok0 = (lm < 12);
    const bool rowok1 = (lm < 6);
    const int  erow1  = rowok1 ? lm : 0;
    const bool colok  = (lm < 8);
    const int  qcol   = lm & 7;

    for (int ch = 0; ch < nch; ++ch) {
        int t = tstart + (ch << 5) + lane;
        float rad[6], bang[8];
        int cls = -1;
        if (t < tend) {
            int j = idx_j[t], k = idx_k[t];
            float e0 = est[t * 3 + 0], e1 = est[t * 3 + 1], e2 = est[t * 3 + 2];
            float sx = e0 * cell[cb + 0] + e1 * cell[cb + 3] + e2 * cell[cb + 6];
            float sy = e0 * cell[cb + 1] + e1 * cell[cb + 4] + e2 * cell[cb + 7];
            float sz = e0 * cell[cb + 2] + e1 * cell[cb + 5] + e2 * cell[cb + 8];
            float vjx = pos[j * 3 + 0] - xi + sx;
            float vjy = pos[j * 3 + 1] - yi + sy;
            float vjz = pos[j * 3 + 2] - zi + sz;
            float vkx = pos[k * 3 + 0] - xi + sx;
            float vky = pos[k * 3 + 1] - yi + sy;
            float vkz = pos[k * 3 + 2] - zi + sz;
            float rij2 = vjx * vjx + vjy * vjy + vjz * vjz;
            float rik2 = vkx * vkx + vky * vky + vkz * vkz;
            float dxx = vkx - vjx, dyy = vky - vjy, dzz = vkz - vjz;
            float rjk2 = dxx * dxx + dyy * dyy + dzz * dzz;
            float rij = sqrtf(rij2), rik = sqrtf(rik2), rjk = sqrtf(rjk2);
            float cosv = (vjx * vkx + vjy * vky + vjz * vkz) / (rij * rik + 1e-12f);
            float ssum = (rij2 + rik2 + rjk2) * INV_RC2;
#pragma unroll
            for (int e = 0; e < 6; ++e) rad[e] = __expf(-eta4[e] * ssum);
            float fc3 = fcut(rij) * fcut(rik) * fcut(rjk);
            float up = 1.0f + cosv, um = 1.0f - cosv;
            float up2 = up * up,  um2 = um * um;
            float up4 = up2 * up2, um4 = um2 * um2;
            float up8 = up4 * up4, um8 = um4 * um4;
            bang[0] = fc3 * up;              bang[1] = fc3 * um;
            bang[2] = 0.5f * fc3 * up2;      bang[3] = 0.5f * fc3 * um2;
            bang[4] = 0.125f * fc3 * up4;    bang[5] = 0.125f * fc3 * um4;
            bang[6] = 0.015625f * fc3 * up8; bang[7] = 0.015625f * fc3 * um8;
            int zj = z[j], zk = z[k];
            cls = (zj == 1 && zk == 1) ? 0 : ((zj == 8 && zk == 8) ? 1 : 2);
        } else {
#pragma unroll
            for (int e = 0; e < 6; ++e) rad[e] = 0.0f;
#pragma unroll
            for (int q = 0; q < 8; ++q) bang[q] = 0.0f;
        }
#pragma unroll
        for (int e = 0; e < 6; ++e) s_rad[lane][e] = rad[e];
#pragma unroll
        for (int q = 0; q < 8; ++q) s_bang[lane][q] = bang[q];
        s_cls[lane] = cls;
        __syncthreads();

        // 8 K-steps of 4 triplets each; EXEC is all-ones here (no divergence).
#pragma unroll
        for (int kk = 0; kk < 8; ++kk) {
            int t0 = (kk << 2) + (hf << 1);
            int t1 = t0 + 1;
            v2f fa0, fa1, fb;
            fa0.x = (rowok0 && s_cls[t0] == srow0) ? s_rad[t0][erow0] : 0.0f;
            fa0.y = (rowok0 && s_cls[t1] == srow0) ? s_rad[t1][erow0] : 0.0f;
            fa1.x = (rowok1 && s_cls[t0] == 2)     ? s_rad[t0][erow1] : 0.0f;
            fa1.y = (rowok1 && s_cls[t1] == 2)     ? s_rad[t1][erow1] : 0.0f;
            fb.x  = colok ? s_bang[t0][qcol] : 0.0f;
            fb.y  = colok ? s_bang[t1][qcol] : 0.0f;
            c0 = __builtin_amdgcn_wmma_f32_16x16x4_f32(false, fa0, false, fb,
                                                       (short)0, c0, false, false);
            c1 = __builtin_amdgcn_wmma_f32_16x16x4_f32(false, fa1, false, fb,
                                                       (short)0, c1, false, false);
        }
        __syncthreads();
    }

    // D layout: VGPR v, lanes 0-15 -> M=v ; lanes 16-31 -> M=v+8 ; N = lm.
    float* orow = out + (size_t)a * NDESC + 12;
#pragma unroll
    for (int v = 0; v < 8; ++v) {
        int M = hf ? (v + 8) : v;
        if (M < 12 && lm < 8) {
            int s = M / 6, e = M - s * 6;
            orow[s * 48 + e * 8 + lm] = 0.5f * c0[v];
        }
        if (M < 6 && lm < 8) {
            orow[96 + M * 8 + lm] = 0.5f * c1[v];
        }
    }
}

// ---------------------------------------------------------------- launcher
extern "C" void kernel_launch(void* const* d_in, const int* in_sizes, int n_in,
                              void* d_out, int out_size, void* d_ws, size_t ws_size,
                              hipStream_t stream) {
    const float* pos    = (const float*)d_in[0];
    const float* cell   = (const float*)d_in[1];
    const float* eshift = (const float*)d_in[2];
    const float* est    = (const float*)d_in[3];
    const float* mean   = (const float*)d_in[4];
    const float* stdv   = (const float*)d_in[5];
    const int*   z      = (const int*)d_in[6];
    const int*   eidx   = (const int*)d_in[7];
    const int*   batch  = (const int*)d_in[8];
    const int*   idx_i  = (const int*)d_in[9];
    const int*   idx_j  = (const int*)d_in[10];
    const int*   idx_k  = (const int*)d_in[11];
    float* out = (float*)d_out;

    const int N     = in_sizes[6];      // atoms
    const int E     = in_sizes[7] / 2;  // edges
    const int n_tri = in_sizes[9];      // triplets

    zero_kernel<<<(out_size + 255) / 256, 256, 0, stream>>>(out, out_size);
    g2_kernel<<<(E + 255) / 256, 256, 0, stream>>>(pos, cell, eshift, z, batch,
                                                   eidx, E, out);
    g4_kernel<<<N, 32, 0, stream>>>(pos, cell, est, z, batch,
                                    idx_i, idx_j, idx_k, n_tri, out);
    norm_kernel<<<(out_size + 255) / 256, 256, 0, stream>>>(out, mean, stdv,
                                                            out_size);
}